// TaylorFeatureMap_26809185861989
// MI455X (gfx1250) — compile-verified
//
#include <hip/hip_runtime.h>

// CDNA5 / gfx1250: wave32. One wave computes z = x xT for one 16-dim row via
// V_WMMA_F32_16X16X4_F32 (fp32-exact rank-1 outer product), stages the
// 153-float feature row in LDS (branch-free: address/value selects only),
// and the whole block streams 8 consecutive rows out with b128 stores.
// Ragged tails are handled by overlapping the last group (rewrites identical
// values) so there is exactly ONE store path -> no static code bloat.
typedef __attribute__((ext_vector_type(2))) float v2f;
typedef __attribute__((ext_vector_type(8))) float v8f;

#define HEAD_DIM        16
#define OUT_DIM         153          // 1 + 16 + 16 + 120
#define WAVES_PER_BLOCK 8
#define THREADS         256
#define BLOCK_ROWS      8            // one row per wave per block iteration
#define ITERS_PER_BLOCK 8            // target loop trips per block
#define GROUP_FLOATS    (BLOCK_ROWS * OUT_DIM)   // 1224 floats = 306 float4

__global__ __launch_bounds__(THREADS)
void taylor_feature_map_kernel(const float* __restrict__ x,
                               float* __restrict__ out,
                               int nRows, int nGroups)
{
    // 8 rows staged contiguously (1224 floats) + per-wave dummy scribble slots
    __shared__ __align__(16) float lds[GROUP_FLOATS + WAVES_PER_BLOCK];

    const int tid  = threadIdx.x;
    const int lane = tid & 31;
    const int wave = tid >> 5;

    // d = 16 : d^0.25 = 2 ; sqrt(d) = 4
    const float inv_rrd = 0.5f;                              // 1 / d^0.25
    const float inv_rd  = 0.25f;                             // 1 / sqrt(d)
    const float diag_sc = 0.25f * 0.70710678118654752440f;   // 1 / (sqrt(d)*sqrt(2))

    const int dummy = GROUP_FLOATS + wave;                   // masked-lane sink

    // Loop-invariant: out[row][0] = 1.0. Each wave writes its slot once; the
    // streamed region is re-staged but this slot's value never changes.
    {
        const int adr0 = (lane == 0) ? wave * OUT_DIM : dummy;
        lds[adr0] = 1.0f;
    }

    for (int g = blockIdx.x; g < nGroups; g += gridDim.x) {
        // Full 8-row group always; ragged last group overlaps the previous
        // one and rewrites identical values (nRows % 8 == 0 here -> no-op).
        const int base = min(g * BLOCK_ROWS, nRows - BLOCK_ROWS);
        const int row  = base + wave;                        // always < nRows

        // ---- unconditional 64B row load (lanes 16-31 duplicate, L0 hit)
        const float xv = x[(size_t)row * HEAD_DIM + (lane & 15)];

        // A (16x4) / B (4x16) f32: only K=0 populated; lanes 16-31 and
        // VGPR1 zeroed so D = x xT exactly, independent of K striping.
        const float xa = (lane < 16) ? xv : 0.0f;            // v_cndmask
        v2f a; a.x = xa; a.y = 0.0f;
        v8f c = {};
        c = __builtin_amdgcn_wmma_f32_16x16x4_f32(
                false, a, false, a, (short)0, c, false, false);

        float* wl = lds + wave * OUT_DIM;

        // out[1..16] = x/2 (lanes 0-15) ; out[17..32] = x^2*sc (lanes 16-31)
        // both map to LDS index 1+lane -> single unconditional ds_store
        wl[1 + lane] = (lane < 16) ? xv * inv_rrd : xv * xv * diag_sc;

        // C/D layout: VGPR j, lane l -> z[m][n], m = j + 8*(l>=16), n = l&15
        // triu (m<n) row-major index: t = 15m - m(m-1)/2 + (n-m-1)
        // Addresses are loop-invariant per lane -> hoisted by the compiler.
        const int m_base = (lane >> 4) << 3;
        const int n      = lane & 15;
        #pragma unroll
        for (int j = 0; j < 8; ++j) {
            const int m = m_base + j;
            const int t = 15 * m - ((m * (m - 1)) >> 1) + (n - m - 1);
            const int ad = (n > m) ? (wave * OUT_DIM + 33 + t) : dummy;
            lds[ad] = c[j] * inv_rd;                         // z_ij / sqrt(d)
        }
        __syncthreads();

        // ---- block-cooperative stream-out: 306 float4 = 4896 B contiguous.
        // base multiple of 8 -> dst 16B-aligned (8*612 % 16 == 0).
        float4*       dst4 = (float4*)(out + (size_t)base * OUT_DIM);
        const float4* src4 = (const float4*)lds;
        dst4[tid] = src4[tid];                               // 256 float4
        if (tid < GROUP_FLOATS / 4 - THREADS)                // remaining 50
            dst4[THREADS + tid] = src4[THREADS + tid];
        __syncthreads();   // protect LDS against next iteration's staging (WAR)
    }
}

extern "C" void kernel_launch(void* const* d_in, const int* in_sizes, int n_in,
                              void* d_out, int out_size, void* d_ws, size_t ws_size,
                              hipStream_t stream) {
    (void)n_in; (void)out_size; (void)d_ws; (void)ws_size;

    const float* x  = (const float*)d_in[0];
    float*      out = (float*)d_out;

    const int nRows   = in_sizes[0] / HEAD_DIM;              // 524288
    const int nGroups = (nRows + BLOCK_ROWS - 1) / BLOCK_ROWS; // 65536
    int blocks = (nGroups + ITERS_PER_BLOCK - 1) / ITERS_PER_BLOCK; // 8192
    if (blocks < 1) blocks = 1;

    taylor_feature_map_kernel<<<blocks, THREADS, 0, stream>>>(x, out, nRows, nGroups);
}